// QRNN_61942018343372
// MI455X (gfx1250) — compile-verified
//
#include <hip/hip_runtime.h>
#include <hip/hip_bf16.h>

// ---------------------------------------------------------------------------
// QRNN: angles = x @ W.T + b  (WMMA f32 GEMM, 65536x512 @ 512x69)
// then 6-qubit / 3-layer circuit simulation per batch row (VALU, in-register
// statevector), output (B, 2, 64) float32.
// ---------------------------------------------------------------------------

typedef __attribute__((ext_vector_type(2))) float v2f;
typedef __attribute__((ext_vector_type(8))) float v8f;

#define NQ       6
#define NSTATES  64
#define NLAYER   3
#define PER_LAYER 23          // 4*6 - 1
#define NANG     69           // 3 * 23
#define KDIM     512
#define ROWS     128          // batch rows per block (= threads per block)
#define NPAD     71           // odd stride -> conflict-free per-thread row reads

// ---------------- circuit gates (compile-time wires, fully unrolled) --------

template <int W0, int W1>
__device__ __forceinline__ void rxx_gate(float* re, float* im, float th) {
    float s, c;
    __sincosf(0.5f * th, &s, &c);
    constexpr int B0 = 1 << (5 - W0);
    constexpr int B1 = 1 << (5 - W1);
    constexpr int MK = B0 | B1;
#pragma unroll
    for (int p = 0; p < NSTATES; ++p) {
        if (p & B0) continue;          // visit each {p, p^MK} pair once
        const int q = p ^ MK;
        const float pr = re[p], pi = im[p];
        const float qr = re[q], qi = im[q];
        // new = c*old[p] + (-i*s)*old[p^mask]
        re[p] = fmaf(c, pr,  s * qi);
        im[p] = fmaf(c, pi, -s * qr);
        re[q] = fmaf(c, qr,  s * pi);
        im[q] = fmaf(c, qi, -s * pr);
    }
}

template <int WQ>
__device__ __forceinline__ void u3_gate(float* re, float* im,
                                        float th, float ph, float lm) {
    float st, ct;  __sincosf(0.5f * th, &st, &ct);
    float sp, cp;  __sincosf(ph,        &sp, &cp);
    float sl, cl;  __sincosf(lm,        &sl, &cl);
    float spl, cpl; __sincosf(ph + lm,  &spl, &cpl);
    // U = [[ct, -e^{i lm} st], [e^{i ph} st, e^{i (ph+lm)} ct]]
    const float u01r = -cl * st, u01i = -sl * st;
    const float u10r =  cp * st, u10i =  sp * st;
    const float u11r = cpl * ct, u11i = spl * ct;
    constexpr int BW = 1 << (5 - WQ);
#pragma unroll
    for (int p = 0; p < NSTATES; ++p) {
        if (p & BW) continue;
        const int q = p | BW;
        const float ar = re[p], ai = im[p];
        const float br = re[q], bi = im[q];
        // new_p = ct*a + u01*b
        re[p] = fmaf(ct, ar, fmaf(u01r, br, -u01i * bi));
        im[p] = fmaf(ct, ai, fmaf(u01r, bi,  u01i * br));
        // new_q = u10*a + u11*b
        re[q] = fmaf(u10r, ar, fmaf(-u10i, ai, fmaf(u11r, br, -u11i * bi)));
        im[q] = fmaf(u10i, ar, fmaf( u10r, ai, fmaf(u11i, br,  u11r * bi)));
    }
}

// ---------------- fused kernel ---------------------------------------------

__global__ __launch_bounds__(ROWS)
void qrnn_fused_kernel(const float* __restrict__ x,
                       const float* __restrict__ W,
                       const float* __restrict__ b,
                       float* __restrict__ out) {
    __shared__ float s_ang[ROWS * NPAD];

    const int tid  = threadIdx.x;
    const int lane = tid & 31;
    const int wave = tid >> 5;
    const int row0 = blockIdx.x * ROWS;

    const int l16 = lane & 15;
    const int kh  = (lane >> 4) << 1;   // K sub-offset: 0 or 2

    // ---- Phase 1: angles[128 x 69] = x_tile @ W.T + b via WMMA f32 ----
    // 8 mtiles x 5 ntiles = 40 16x16 tiles; 4 waves -> 10 tiles each.
#pragma unroll 1
    for (int t = 0; t < 10; ++t) {
        const int tile = wave * 10 + t;
        const int mt = tile / 5;
        const int nt = tile % 5;

        const int m = row0 + mt * 16 + l16;
        const int n = nt * 16 + l16;
        const bool nok = (n < NANG);
        const int nc = nok ? n : (NANG - 1);

        const float* xp = x + (size_t)m * KDIM + kh;
        const float* wp = W + (size_t)nc * KDIM + kh;

        v8f acc = {};
#pragma unroll 8
        for (int k = 0; k < KDIM; k += 4) {
            v2f av = *(const v2f*)(xp + k);
            v2f bv = *(const v2f*)(wp + k);
            if (!nok) { bv[0] = 0.0f; bv[1] = 0.0f; }   // mask pad columns
            // D = A(16x4 f32) * B(4x16 f32) + C, full fp32
            acc = __builtin_amdgcn_wmma_f32_16x16x4_f32(
                      false, av, false, bv, (short)0, acc, false, false);
        }

        if (nok) {
            const float bias = b[n];
            const int mb = mt * 16 + ((lane >> 4) << 3);  // VGPR e -> row mb+e
#pragma unroll
            for (int e = 0; e < 8; ++e)
                s_ang[(mb + e) * NPAD + n] = acc[e] + bias;
        }
    }
    __syncthreads();

    // ---- Phase 2: per-thread 6-qubit circuit, state in registers ----
    const float* a = &s_ang[tid * NPAD];

    float re[NSTATES], im[NSTATES];
#pragma unroll
    for (int p = 0; p < NSTATES; ++p) { re[p] = 0.0f; im[p] = 0.0f; }
    re[0] = 1.0f;

#pragma unroll
    for (int lay = 0; lay < NLAYER; ++lay) {
        const int base = lay * PER_LAYER;
        int k = base;
        rxx_gate<0, 1>(re, im, a[k]);
        u3_gate<0>(re, im, a[k + 1], a[k + 2], a[k + 3]);  k += 4;
        rxx_gate<1, 2>(re, im, a[k]);
        u3_gate<1>(re, im, a[k + 1], a[k + 2], a[k + 3]);  k += 4;
        rxx_gate<2, 3>(re, im, a[k]);
        u3_gate<2>(re, im, a[k + 1], a[k + 2], a[k + 3]);  k += 4;
        rxx_gate<3, 4>(re, im, a[k]);
        u3_gate<3>(re, im, a[k + 1], a[k + 2], a[k + 3]);  k += 4;
        rxx_gate<4, 5>(re, im, a[k]);
        u3_gate<4>(re, im, a[k + 1], a[k + 2], a[k + 3]);  k += 4;
        u3_gate<5>(re, im, a[k], a[k + 1], a[k + 2]);
    }

    // ---- Phase 3: write (B, 2, 64) output, float4 stores ----
    float* o = out + (size_t)(row0 + tid) * (2 * NSTATES);
#pragma unroll
    for (int j = 0; j < NSTATES; j += 4) {
        float4 vr = make_float4(re[j], re[j + 1], re[j + 2], re[j + 3]);
        *(float4*)(o + j) = vr;
    }
#pragma unroll
    for (int j = 0; j < NSTATES; j += 4) {
        float4 vi = make_float4(im[j], im[j + 1], im[j + 2], im[j + 3]);
        *(float4*)(o + NSTATES + j) = vi;
    }
}

// ---------------- launch ----------------------------------------------------

extern "C" void kernel_launch(void* const* d_in, const int* in_sizes, int n_in,
                              void* d_out, int out_size, void* d_ws, size_t ws_size,
                              hipStream_t stream) {
    const float* x = (const float*)d_in[0];   // [B, 512]
    const float* W = (const float*)d_in[1];   // [69, 512]
    const float* b = (const float*)d_in[2];   // [69]
    float* out = (float*)d_out;               // [B, 2, 64]

    const int batch = in_sizes[0] / KDIM;     // 65536
    const int grid = batch / ROWS;            // 512

    qrnn_fused_kernel<<<dim3(grid), dim3(ROWS), 0, stream>>>(x, W, b, out);
}